// CQAttention_31387620999414
// MI455X (gfx1250) — compile-verified
//
#include <hip/hip_runtime.h>
#include <math.h>

#define B_   32
#define D_   128
#define LC_  2048
#define LQ_  256

typedef __attribute__((ext_vector_type(2))) float v2f;
typedef __attribute__((ext_vector_type(8))) float v8f;

__device__ __forceinline__ v8f wmma_f32(v2f a, v2f b, v8f c) {
  // V_WMMA_F32_16X16X4_F32 : D = A(16x4) x B(4x16) + C(16x16), fp32 all the way
  return __builtin_amdgcn_wmma_f32_16x16x4_f32(
      /*neg_a=*/false, a, /*neg_b=*/false, b,
      /*c_mod=*/(short)0, c, /*reuse_a=*/false, /*reuse_b=*/false);
}

// ---------------------------------------------------------------------------
// Rank-1 terms: csum[b,c] = sum_d C[b,d,c]*w4C[d] ; qsum[b,q] = sum_d Q[b,d,q]*w4Q[d]
// ---------------------------------------------------------------------------
__global__ void k_csum(const float* __restrict__ C, const float* __restrict__ w4C,
                       float* __restrict__ csum) {
  int idx = blockIdx.x * blockDim.x + threadIdx.x;        // [0, B*LC)
  int b = idx / LC_, c = idx % LC_;
  const float* Cb = C + (size_t)b * D_ * LC_ + c;
  float s = 0.f;
#pragma unroll 8
  for (int d = 0; d < D_; ++d) s += Cb[(size_t)d * LC_] * w4C[d];
  csum[idx] = s;
}

__global__ void k_qsum(const float* __restrict__ Q, const float* __restrict__ w4Q,
                       float* __restrict__ qsum) {
  int idx = blockIdx.x * blockDim.x + threadIdx.x;        // [0, B*LQ)
  int b = idx / LQ_, q = idx % LQ_;
  const float* Qb = Q + (size_t)b * D_ * LQ_ + q;
  float s = 0.f;
#pragma unroll 8
  for (int d = 0; d < D_; ++d) s += Qb[(size_t)d * LQ_] * w4Q[d];
  qsum[idx] = s;
}

// ---------------------------------------------------------------------------
// S[b,c,q] = (Ct*w4mlu) @ Qt^T + csum[c] + qsum[q] + bias     (one wave / 16x16 tile)
// ---------------------------------------------------------------------------
__global__ void __launch_bounds__(32)
k_similarity(const float* __restrict__ C, const float* __restrict__ Q,
             const float* __restrict__ w4mlu, const float* __restrict__ biasp,
             const float* __restrict__ csum, const float* __restrict__ qsum,
             float* __restrict__ S) {
  const int lane = threadIdx.x;
  const int q0 = blockIdx.x * 16;
  const int c0 = blockIdx.y * 16;
  const int b  = blockIdx.z;
  const int ln = lane & 15;              // M/N index within tile
  const int kh = (lane >> 4) << 1;       // K half: 0 or 2
  const float bias = biasp[0];
  const float* Cb = C + (size_t)b * D_ * LC_;
  const float* Qb = Q + (size_t)b * D_ * LQ_;
  v8f acc = {};
#pragma unroll 4
  for (int d0 = 0; d0 < D_; d0 += 4) {
    const int dk = d0 + kh;
    v2f a, bf;
    a.x  = Cb[(size_t)(dk + 0) * LC_ + c0 + ln] * w4mlu[dk + 0];
    a.y  = Cb[(size_t)(dk + 1) * LC_ + c0 + ln] * w4mlu[dk + 1];
    bf.x = Qb[(size_t)(dk + 0) * LQ_ + q0 + ln];
    bf.y = Qb[(size_t)(dk + 1) * LQ_ + q0 + ln];
    acc = wmma_f32(a, bf, acc);
  }
  const int rowOff = (lane >> 4) << 3;   // M offset: 0 or 8
  const float qs = qsum[b * LQ_ + q0 + ln];
  float* Sb = S + ((size_t)b * LC_ + c0) * LQ_ + q0;
#pragma unroll
  for (int r = 0; r < 8; ++r) {
    const int cm = rowOff + r;
    Sb[(size_t)cm * LQ_ + ln] = acc[r] + csum[b * LC_ + c0 + cm] + qs + bias;
  }
}

// ---------------------------------------------------------------------------
// Row softmax stats over q (masked by Qmask): one wave per (b,c) row
// ---------------------------------------------------------------------------
__global__ void k_rowstats(const float* __restrict__ S, const int* __restrict__ Qmask,
                           float* __restrict__ rmax, float* __restrict__ rsumInv) {
  const int wave = blockIdx.x * (blockDim.x >> 5) + (threadIdx.x >> 5);  // flat b*LC+c
  const int lane = threadIdx.x & 31;
  const int b = wave / LC_;
  const float* row = S + (size_t)wave * LQ_;
  const int* qm = Qmask + b * LQ_;
  float vals[8]; int msk[8];
  float m = -INFINITY;
#pragma unroll
  for (int i = 0; i < 8; ++i) {
    const int q = lane + 32 * i;
    vals[i] = row[q];
    msk[i]  = qm[q];
    if (msk[i]) m = fmaxf(m, vals[i]);
  }
#pragma unroll
  for (int off = 16; off > 0; off >>= 1) m = fmaxf(m, __shfl_xor(m, off, 32));
  float s = 0.f;
#pragma unroll
  for (int i = 0; i < 8; ++i)
    if (msk[i]) s += __expf(vals[i] - m);
#pragma unroll
  for (int off = 16; off > 0; off >>= 1) s += __shfl_xor(s, off, 32);
  if (lane == 0) {
    rmax[wave]    = m;
    rsumInv[wave] = 1.f / s;
  }
}

// ---------------------------------------------------------------------------
// Column softmax stats over c (masked by Cmask): thread-per-q, online softmax
// ---------------------------------------------------------------------------
__global__ void k_colstats(const float* __restrict__ S, const int* __restrict__ Cmask,
                           float* __restrict__ cmax, float* __restrict__ csumInv) {
  const int b = blockIdx.x;
  const int q = threadIdx.x;             // 256 threads
  const float* Sb = S + (size_t)b * LC_ * LQ_ + q;
  const int* cm = Cmask + b * LC_;
  float m = -INFINITY, s = 0.f;
  for (int c = 0; c < LC_; ++c) {
    if (!cm[c]) continue;
    const float v = Sb[(size_t)c * LQ_];
    const float nm = fmaxf(m, v);
    s = s * __expf(m - nm) + __expf(v - nm);
    m = nm;
  }
  cmax[b * LQ_ + q]    = m;
  csumInv[b * LQ_ + q] = 1.f / s;
}

// ---------------------------------------------------------------------------
// T[b,q,d] = (S2^T @ Ct)[q,d] : M=q, N=d, K=c.  S2 reconstructed on the fly.
// ---------------------------------------------------------------------------
__global__ void __launch_bounds__(32)
k_T(const float* __restrict__ S, const float* __restrict__ C,
    const int* __restrict__ Cmask, const float* __restrict__ cmax,
    const float* __restrict__ csumInv, float* __restrict__ T) {
  const int lane = threadIdx.x;
  const int d0 = blockIdx.x * 16;
  const int q0 = blockIdx.y * 16;
  const int b  = blockIdx.z;
  const int ln = lane & 15;
  const int kh = (lane >> 4) << 1;
  const int qA = q0 + ln;                // A-frag row (fixed per lane)
  const float mA = cmax[b * LQ_ + qA];
  const float iA = csumInv[b * LQ_ + qA];
  const float* Sb  = S + (size_t)b * LC_ * LQ_;
  const float* Cb  = C + (size_t)b * D_ * LC_ + (size_t)(d0 + ln) * LC_;
  const int*   cmk = Cmask + b * LC_;
  v8f acc = {};
  for (int cb = 0; cb < LC_; cb += 4) {
    const int ck = cb + kh;
    v2f a, bf;
    const float s0 = Sb[(size_t)(ck + 0) * LQ_ + qA];
    const float s1 = Sb[(size_t)(ck + 1) * LQ_ + qA];
    a.x  = cmk[ck + 0] ? __expf(s0 - mA) * iA : 0.f;
    a.y  = cmk[ck + 1] ? __expf(s1 - mA) * iA : 0.f;
    bf.x = Cb[ck + 0];
    bf.y = Cb[ck + 1];
    acc = wmma_f32(a, bf, acc);
  }
  const int rowOff = (lane >> 4) << 3;
  float* Tb = T + (size_t)b * LQ_ * D_;
#pragma unroll
  for (int r = 0; r < 8; ++r) {
    const int q = q0 + rowOff + r;
    Tb[(size_t)q * D_ + d0 + ln] = acc[r];
  }
}

// ---------------------------------------------------------------------------
// Final: A = S1@Qt, Bt = S1@T, emit [Ct, A, Ct*A, Ct*Bt] -> out[b, 4D, Lc]
// One wave per 16-wide c-tile; LDS transpose for coalesced c-major stores.
// ---------------------------------------------------------------------------
__global__ void __launch_bounds__(32)
k_out(const float* __restrict__ C, const float* __restrict__ Q,
      const float* __restrict__ S, const float* __restrict__ T,
      const int* __restrict__ Qmask,
      const float* __restrict__ rmax, const float* __restrict__ rsumInv,
      float* __restrict__ out) {
  __shared__ float ldsC[D_][16];
  __shared__ float ldsA[D_][16];
  __shared__ float ldsB[D_][16];
  const int lane = threadIdx.x;
  const int c0 = blockIdx.x * 16;
  const int b  = blockIdx.y;
  const int ln = lane & 15;
  const int kh = (lane >> 4) << 1;
  const int rowOff = (lane >> 4) << 3;
  const int cA = c0 + ln;                // A-frag row (fixed per lane)
  const float mA = rmax[b * LC_ + cA];
  const float iA = rsumInv[b * LC_ + cA];
  const float* Sb = S + ((size_t)b * LC_ + cA) * LQ_;
  const float* Qb = Q + (size_t)b * D_ * LQ_;
  const float* Tb = T + (size_t)b * LQ_ * D_;
  const int*   qm = Qmask + b * LQ_;

  v8f accA[8], accB[8];
#pragma unroll
  for (int dt = 0; dt < 8; ++dt) {
    accA[dt] = (v8f)(0.0f);
    accB[dt] = (v8f)(0.0f);
  }

  for (int q0 = 0; q0 < LQ_; q0 += 4) {
    const int qk = q0 + kh;
    v2f a;
    const float s0 = Sb[qk + 0];
    const float s1 = Sb[qk + 1];
    a.x = qm[qk + 0] ? __expf(s0 - mA) * iA : 0.f;
    a.y = qm[qk + 1] ? __expf(s1 - mA) * iA : 0.f;
#pragma unroll
    for (int dt = 0; dt < 8; ++dt) {
      const int d = dt * 16 + ln;
      v2f bq, bt;
      bq.x = Qb[(size_t)d * LQ_ + qk + 0];
      bq.y = Qb[(size_t)d * LQ_ + qk + 1];
      accA[dt] = wmma_f32(a, bq, accA[dt]);
      bt.x = Tb[(size_t)(qk + 0) * D_ + d];
      bt.y = Tb[(size_t)(qk + 1) * D_ + d];
      accB[dt] = wmma_f32(a, bt, accB[dt]);
    }
  }

  // Stage C tile (coalesced) and WMMA results into LDS as [d][c]
  const float* Cb = C + (size_t)b * D_ * LC_ + c0;
  for (int i = lane; i < D_ * 16; i += 32) {
    const int d = i >> 4, cc = i & 15;
    ldsC[d][cc] = Cb[(size_t)d * LC_ + cc];
  }
#pragma unroll
  for (int dt = 0; dt < 8; ++dt) {
    const int d = dt * 16 + ln;
#pragma unroll
    for (int r = 0; r < 8; ++r) {
      ldsA[d][rowOff + r] = accA[dt][r];
      ldsB[d][rowOff + r] = accB[dt][r];
    }
  }
  __syncthreads();

  // out[b, f, c0+cc], f in [0,512): sections [Ct, A, Ct*A, Ct*Bt]
  float* outb = out + (size_t)b * (4 * D_) * LC_ + c0;
  const int fhalf = lane >> 4;
  for (int f2 = 0; f2 < 256; ++f2) {
    const int f = f2 * 2 + fhalf;
    const int dd = f & (D_ - 1);
    const int sec = f >> 7;
    const float cv = ldsC[dd][ln];
    const float av = ldsA[dd][ln];
    const float bv = ldsB[dd][ln];
    const float val = (sec == 0) ? cv : (sec == 1) ? av : (sec == 2) ? cv * av : cv * bv;
    outb[(size_t)f * LC_ + ln] = val;
  }
}

// ---------------------------------------------------------------------------
extern "C" void kernel_launch(void* const* d_in, const int* in_sizes, int n_in,
                              void* d_out, int out_size, void* d_ws, size_t ws_size,
                              hipStream_t stream) {
  const float* C     = (const float*)d_in[0];
  const float* Q     = (const float*)d_in[1];
  const int*   Cmask = (const int*)d_in[2];
  const int*   Qmask = (const int*)d_in[3];
  const float* w4C   = (const float*)d_in[4];
  const float* w4Q   = (const float*)d_in[5];
  const float* w4mlu = (const float*)d_in[6];
  const float* bias  = (const float*)d_in[7];
  float* out = (float*)d_out;

  float* ws = (float*)d_ws;
  size_t off = 0;
  float* S        = ws + off; off += (size_t)B_ * LC_ * LQ_;   // 64 MB
  float* csum     = ws + off; off += (size_t)B_ * LC_;
  float* qsum     = ws + off; off += (size_t)B_ * LQ_;
  float* rmaxA    = ws + off; off += (size_t)B_ * LC_;
  float* rsumInvA = ws + off; off += (size_t)B_ * LC_;
  float* cmaxA    = ws + off; off += (size_t)B_ * LQ_;
  float* csumInvA = ws + off; off += (size_t)B_ * LQ_;
  float* T        = ws + off; off += (size_t)B_ * LQ_ * D_;    // 4 MB

  k_csum<<<(B_ * LC_) / 256, 256, 0, stream>>>(C, w4C, csum);
  k_qsum<<<(B_ * LQ_) / 256, 256, 0, stream>>>(Q, w4Q, qsum);
  k_similarity<<<dim3(LQ_ / 16, LC_ / 16, B_), 32, 0, stream>>>(
      C, Q, w4mlu, bias, csum, qsum, S);
  k_rowstats<<<(B_ * LC_) / 8, 256, 0, stream>>>(S, Qmask, rmaxA, rsumInvA);
  k_colstats<<<B_, 256, 0, stream>>>(S, Cmask, cmaxA, csumInvA);
  k_T<<<dim3(D_ / 16, LQ_ / 16, B_), 32, 0, stream>>>(
      S, C, Cmask, cmaxA, csumInvA, T);
  k_out<<<dim3(LC_ / 16, B_), 32, 0, stream>>>(
      C, Q, S, T, Qmask, rmaxA, rsumInvA, out);
}